// RBF_Kernel_44865228374551
// MI455X (gfx1250) — compile-verified
//
#include <hip/hip_runtime.h>
#include <cstddef>

typedef __attribute__((ext_vector_type(2))) float v2f;
typedef __attribute__((ext_vector_type(8))) float v8f;

#define N_ROWS 8192
#define M_ROWS 8192
#define DDIM   64
#define LIM    36.0f

// ---------------- workspace layout (floats) ----------------
// w      : [0,    64)
// sigma  : [64,   65)   (padded to 128)
// x2     : [128,  128+8192)
// z2     : [8320, 8320+8192)
// Xs     : [16512, 16512+8192*64)
// Zs     : [540800, 540800+8192*64)
#define WS_W     0
#define WS_SIGMA 64
#define WS_X2    128
#define WS_Z2    8320
#define WS_XS    16512
#define WS_ZS    540800

__device__ __forceinline__ float softplus_f(float x) {
    return (x > LIM) ? x : log1pf(__expf(fminf(x, LIM)));
}

// ---- kernel 1: hyperparameter transform (1 block, 64 threads) ----
__global__ void rbf_params(const float* __restrict__ l_raw,
                           const float* __restrict__ sigma_raw,
                           float* __restrict__ w,
                           float* __restrict__ sigma) {
    int t = threadIdx.x;
    if (t < DDIM) {
        float l = softplus_f(l_raw[t]);
        w[t] = 1.0f / l;                 // sqrt(1/l^2) == 1/l
    }
    if (t == 0) {
        sigma[0] = softplus_f(sigma_raw[0]);
    }
}

// ---- kernel 2: scale rows by w, compute squared norms ----
__global__ __launch_bounds__(256) void rbf_scale(const float* __restrict__ X,
                                                 const float* __restrict__ Z,
                                                 const float* __restrict__ w,
                                                 float* __restrict__ Xs,
                                                 float* __restrict__ Zs,
                                                 float* __restrict__ x2,
                                                 float* __restrict__ z2) {
    int id = blockIdx.x * blockDim.x + threadIdx.x;     // 0 .. 16383
    const float* src;
    float* dst;
    float* nrm;
    int r;
    if (id < N_ROWS) { src = X; dst = Xs; nrm = x2; r = id; }
    else             { src = Z; dst = Zs; nrm = z2; r = id - N_ROWS; }

    const float4* srow = (const float4*)(src + (size_t)r * DDIM);
    float4*       drow = (float4*)(dst + (size_t)r * DDIM);
    const float4* wv   = (const float4*)w;

    float s = 0.0f;
#pragma unroll
    for (int i = 0; i < DDIM / 4; ++i) {
        float4 v  = srow[i];
        float4 ww = wv[i];
        v.x *= ww.x; v.y *= ww.y; v.z *= ww.z; v.w *= ww.w;
        drow[i] = v;
        s += v.x * v.x + v.y * v.y + v.z * v.z + v.w * v.w;
    }
    nrm[r] = s;
}

// ---- kernel 3: WMMA GEMM + RBF epilogue ----
// Block = 256 threads = 8 waves. Block tile = 128 (M) x 128 (N), full K = 64.
// Each wave computes a 16x128 strip: 8 accumulators of 16x16, K-loop of 16
// V_WMMA_F32_16X16X4_F32 steps each.
#define BM 128
#define BN 128
#define LDP 68                 // LDS row pitch in floats (bank-conflict-free, 16B aligned)

__global__ __launch_bounds__(256) void rbf_main(const float* __restrict__ Xs,
                                                const float* __restrict__ Zs,
                                                const float* __restrict__ x2,
                                                const float* __restrict__ z2,
                                                const float* __restrict__ sig,
                                                float* __restrict__ out) {
    __shared__ float smemA[BM * LDP];
    __shared__ float smemB[BN * LDP];

    const int tid    = threadIdx.x;
    const int lane   = tid & 31;
    const int wave   = tid >> 5;
    const int laneLo = lane & 15;
    const int hi     = lane >> 4;          // 0: K 0..1, 1: K 2..3 within a k-step

    const int rowBlk = blockIdx.y * BM;    // M origin of block tile
    const int colBlk = blockIdx.x * BN;    // N origin of block tile

    // -------- cooperative load of A(128x64) and B(128x64) panels into LDS --------
    // 2048 float4 per panel, 256 threads -> 8 iterations, fully coalesced.
#pragma unroll
    for (int it = 0; it < 8; ++it) {
        int lin = it * 256 + tid;          // float4 index within panel
        int r   = lin >> 4;                // 16 float4 per 64-float row
        int c4  = lin & 15;
        float4 va = *(const float4*)(Xs + (size_t)(rowBlk + r) * DDIM + c4 * 4);
        float4 vb = *(const float4*)(Zs + (size_t)(colBlk + r) * DDIM + c4 * 4);
        *(float4*)(smemA + r * LDP + c4 * 4) = va;
        *(float4*)(smemB + r * LDP + c4 * 4) = vb;
    }
    __syncthreads();

    v8f acc[8];
#pragma unroll
    for (int n = 0; n < 8; ++n) acc[n] = (v8f){0, 0, 0, 0, 0, 0, 0, 0};

    const float* aRow = smemA + (wave * 16 + laneLo) * LDP;

    // -------- K loop: 16 steps of K=4 --------
#pragma unroll
    for (int kk = 0; kk < 16; ++kk) {
        const int kb = kk * 4 + hi * 2;    // ISA A/B layout: lanes<16 hold K{0,1}, lanes>=16 K{2,3}
        v2f a = *(const v2f*)(aRow + kb);
#pragma unroll
        for (int n = 0; n < 8; ++n) {
            v2f b = *(const v2f*)(smemB + (n * 16 + laneLo) * LDP + kb);
            acc[n] = __builtin_amdgcn_wmma_f32_16x16x4_f32(
                /*neg_a=*/false, a, /*neg_b=*/false, b,
                /*c_mod=*/(short)0, acc[n],
                /*reuse_a=*/false, /*reuse_b=*/false);
        }
    }

    // -------- epilogue: out = sigma * exp(acc - 0.5*(x2 + z2)) --------
    const float sigma = sig[0];
    const int rowBase = rowBlk + wave * 16;

    float xh[8];
#pragma unroll
    for (int i = 0; i < 8; ++i)
        xh[i] = x2[rowBase + i + 8 * hi];  // uniform per half-wave -> broadcast load

#pragma unroll
    for (int n = 0; n < 8; ++n) {
        const int col = colBlk + n * 16 + laneLo;
        const float zc = z2[col];
#pragma unroll
        for (int i = 0; i < 8; ++i) {
            const int row = rowBase + i + 8 * hi;       // C/D layout: lanes>=16 -> M+8
            float v = sigma * __expf(acc[n][i] - 0.5f * (xh[i] + zc));
            out[(size_t)row * M_ROWS + col] = v;
        }
    }
}

extern "C" void kernel_launch(void* const* d_in, const int* in_sizes, int n_in,
                              void* d_out, int out_size, void* d_ws, size_t ws_size,
                              hipStream_t stream) {
    const float* X     = (const float*)d_in[0];  // [8192, 64]
    const float* Z     = (const float*)d_in[1];  // [8192, 64]
    const float* l_raw = (const float*)d_in[2];  // [64]
    const float* s_raw = (const float*)d_in[3];  // [1]
    float* out = (float*)d_out;                  // [8192, 8192]

    float* ws    = (float*)d_ws;
    float* w     = ws + WS_W;
    float* sigma = ws + WS_SIGMA;
    float* x2    = ws + WS_X2;
    float* z2    = ws + WS_Z2;
    float* Xs    = ws + WS_XS;
    float* Zs    = ws + WS_ZS;

    // 1. hyperparameter transform
    rbf_params<<<1, 64, 0, stream>>>(l_raw, s_raw, w, sigma);

    // 2. feature scaling + norms (16384 rows total)
    rbf_scale<<<(N_ROWS + M_ROWS) / 256, 256, 0, stream>>>(X, Z, w, Xs, Zs, x2, z2);

    // 3. WMMA GEMM + RBF epilogue: 64 x 64 blocks of 128x128 tiles
    dim3 grid(M_ROWS / BN, N_ROWS / BM);
    rbf_main<<<grid, 256, 0, stream>>>(Xs, Zs, x2, z2, sigma, out);
}